// AdjGNN_37778532335679
// MI455X (gfx1250) — compile-verified
//
#include <hip/hip_runtime.h>
#include <hip/hip_bf16.h>

typedef __attribute__((ext_vector_type(2))) float v2f;
typedef __attribute__((ext_vector_type(8))) float v8f;

#define N_NODES 20000
#define N_EDGES 320000
#define N_GRAPHS 100
#define IN_DIM 74
#define IN_PAD 80
#define HID 256
#define NLAYERS 5
#define M1 1024
#define M2 512
#define NEG_SLOPE 0.01f

// ---------------------------------------------------------------------------
// Zero-pad node features [N,74] -> [N,80]
__global__ void pad_x_kernel(const float* __restrict__ nh, float* __restrict__ xp) {
    int idx = blockIdx.x * 256 + threadIdx.x;
    if (idx >= N_NODES * IN_PAD) return;
    int n = idx / IN_PAD, k = idx % IN_PAD;
    xp[idx] = (k < IN_DIM) ? nh[n * IN_DIM + k] : 0.0f;
}

// Zero-pad embedding weight [74,256] -> [80,256]
__global__ void pad_w_kernel(const float* __restrict__ W, float* __restrict__ Wp) {
    int idx = blockIdx.x * 256 + threadIdx.x;
    if (idx >= IN_PAD * HID) return;
    int k = idx / HID, c = idx % HID;
    Wp[idx] = (k < IN_DIM) ? W[k * HID + c] : 0.0f;
}

// ---------------------------------------------------------------------------
// C[M,Nc] = act(A[M,K] @ W[K,Nc] + bias) — fp32 WMMA 16x16x4.
// Each wave computes a 64x16 C block (4 M-tiles sharing one B fragment):
// one B load pair feeds 4 v_wmma ops -> 6 VMEM loads per 4 WMMAs.
// A lane layout (ISA 7.12.2): lane = m, VGPR0/1 = K{0,1} (lanes 0-15) or
// K{2,3} (lanes 16-31); B mirrors with lane = n. C/D: VGPR j -> row j / j+8.
// Out-of-range A rows are clamp-loaded; their garbage only reaches C rows
// that the row-guarded store never writes, so no masking is needed.
// act: 0 = identity, 1 = LeakyReLU(0.01)
__global__ void wmma_gemm_kernel(const float* __restrict__ A,
                                 const float* __restrict__ W,
                                 const float* __restrict__ bias,
                                 float* __restrict__ C,
                                 int M, int Nc, int K, int act) {
    int lane = threadIdx.x & 31;
    int wid  = threadIdx.x >> 5;
    int ntn   = Nc >> 4;
    int ntm4  = (M + 63) >> 6;           // 64-row blocks
    int tile = blockIdx.x * 8 + wid;
    if (tile >= ntm4 * ntn) return;
    int tm4 = tile / ntn;
    int tn  = tile % ntn;

    int half = lane >> 4;                // 0: lanes 0-15, 1: lanes 16-31
    int sub  = lane & 15;
    int kofs = half * 2;                 // K sub-offset {0,2}
    int col  = tn * 16 + sub;            // B/C column handled by this lane

    const float* Arow[4];
    int rowbase = tm4 * 64 + sub;
#pragma unroll
    for (int t = 0; t < 4; ++t) {
        int r = rowbase + t * 16;
        r = (r < M) ? r : (M - 1);       // clamp: keep loads in-bounds
        Arow[t] = A + (size_t)r * K;
    }

    v8f acc[4] = {{}, {}, {}, {}};
    for (int k = 0; k < K; k += 4) {
        v2f b;
        b.x = W[(size_t)(k + kofs) * Nc + col];
        b.y = W[(size_t)(k + kofs + 1) * Nc + col];
        v2f a0 = *(const v2f*)(Arow[0] + k + kofs);
        v2f a1 = *(const v2f*)(Arow[1] + k + kofs);
        v2f a2 = *(const v2f*)(Arow[2] + k + kofs);
        v2f a3 = *(const v2f*)(Arow[3] + k + kofs);
        acc[0] = __builtin_amdgcn_wmma_f32_16x16x4_f32(false, a0, false, b,
                                                       (short)0, acc[0], false, false);
        acc[1] = __builtin_amdgcn_wmma_f32_16x16x4_f32(false, a1, false, b,
                                                       (short)0, acc[1], false, false);
        acc[2] = __builtin_amdgcn_wmma_f32_16x16x4_f32(false, a2, false, b,
                                                       (short)0, acc[2], false, false);
        acc[3] = __builtin_amdgcn_wmma_f32_16x16x4_f32(false, a3, false, b,
                                                       (short)0, acc[3], false, false);
    }

    float bval = bias[col];
#pragma unroll
    for (int t = 0; t < 4; ++t) {
        int mbase = tm4 * 64 + t * 16 + half * 8;
#pragma unroll
        for (int j = 0; j < 8; ++j) {
            int r = mbase + j;
            if (r < M) {
                float v = acc[t][j] + bval;
                if (act == 1) v = (v > 0.0f) ? v : NEG_SLOPE * v;
                C[(size_t)r * Nc + col] = v;
            }
        }
    }
}

// ---------------------------------------------------------------------------
__device__ __forceinline__ int lower_bound_i32(const int* __restrict__ a, int n, int v) {
    int lo = 0, hi = n;
    while (lo < hi) {
        int mid = (lo + hi) >> 1;
        if (a[mid] < v) lo = mid + 1; else hi = mid;
    }
    return lo;
}

// msgw[n,:] = sw0 * relu( (sum_{e: dst==n} h[src[e],:]) / deg + lb ) + sw1 * x[n,:]
// dst is sorted (reference builds it with repeat(arange)), so each node's edge
// range is found with binary search — deterministic, no float atomics.
__global__ void aggregate_kernel(const float* __restrict__ h,
                                 const float* __restrict__ x,
                                 const int* __restrict__ src,
                                 const int* __restrict__ dst,
                                 const float* __restrict__ lb,
                                 const float* __restrict__ sw,
                                 float* __restrict__ msgw) {
    __shared__ int s_lo, s_hi;
    __shared__ int s_src[256];
    int n = blockIdx.x;
    if (threadIdx.x == 0) {
        s_lo = lower_bound_i32(dst, N_EDGES, n);
        s_hi = lower_bound_i32(dst, N_EDGES, n + 1);
    }
    __syncthreads();
    int lo = s_lo, hi = s_hi;
    int c = threadIdx.x;                 // 256 threads == HID channels
    float acc = 0.0f;
    for (int base = lo; base < hi; base += 256) {
        int chunk = min(256, hi - base);
        if (threadIdx.x < chunk) s_src[threadIdx.x] = src[base + threadIdx.x];
        __syncthreads();
        for (int e = 0; e < chunk; ++e)
            acc += h[(size_t)s_src[e] * HID + c];
        __syncthreads();
    }
    float deg = fmaxf((float)(hi - lo), 1.0f);
    float m = acc / deg + lb[c];
    m = fmaxf(m, 0.0f);
    float w0 = sw[0], w1 = sw[1];
    msgw[(size_t)n * HID + c] = w0 * m + w1 * x[(size_t)n * HID + c];
}

// gmean[g,:] = mean over nodes with graph_id==g of x[n,:]  (graph_id sorted)
__global__ void graph_mean_kernel(const float* __restrict__ x,
                                  const int* __restrict__ gid,
                                  float* __restrict__ gmean) {
    __shared__ int s_lo, s_hi;
    int g = blockIdx.x;
    if (threadIdx.x == 0) {
        s_lo = lower_bound_i32(gid, N_NODES, g);
        s_hi = lower_bound_i32(gid, N_NODES, g + 1);
    }
    __syncthreads();
    int c = threadIdx.x;
    float acc = 0.0f;
    for (int r = s_lo; r < s_hi; ++r)
        acc += x[(size_t)r * HID + c];
    float cnt = fmaxf((float)(s_hi - s_lo), 1.0f);
    gmean[(size_t)g * HID + c] = acc / cnt;
}

// out[g] = h2[g,:] . W3[:,0] + b3
__global__ void final_head_kernel(const float* __restrict__ h2,
                                  const float* __restrict__ W3,
                                  const float* __restrict__ b3,
                                  float* __restrict__ out) {
    __shared__ float red[256];
    int g = blockIdx.x, t = threadIdx.x;
    float s = 0.0f;
    for (int k = t; k < M2; k += 256)
        s += h2[(size_t)g * M2 + k] * W3[k];
    red[t] = s;
    __syncthreads();
    for (int off = 128; off > 0; off >>= 1) {
        if (t < off) red[t] += red[t + off];
        __syncthreads();
    }
    if (t == 0) out[g] = red[0] + b3[0];
}

// ---------------------------------------------------------------------------
static inline int ceil_div(int a, int b) { return (a + b - 1) / b; }

extern "C" void kernel_launch(void* const* d_in, const int* in_sizes, int n_in,
                              void* d_out, int out_size, void* d_ws, size_t ws_size,
                              hipStream_t stream) {
    (void)in_sizes; (void)n_in; (void)out_size; (void)ws_size;
    const float* node_h     = (const float*)d_in[0];
    const int*   src        = (const int*)  d_in[1];
    const int*   dst        = (const int*)  d_in[2];
    const int*   graph_id   = (const int*)  d_in[3];
    const float* node_emb_W = (const float*)d_in[4];
    const float* node_emb_b = (const float*)d_in[5];
    const float* emb_W      = (const float*)d_in[6];   // [L,256,256]
    const float* emb_b      = (const float*)d_in[7];   // [L,256]
    const float* layer_bias = (const float*)d_in[8];   // [L,256]
    const float* sum_w      = (const float*)d_in[9];   // [L,2]
    const float* trans_W    = (const float*)d_in[10];  // [L,256,256]
    const float* trans_b    = (const float*)d_in[11];  // [L,256]
    const float* mlp_W1     = (const float*)d_in[12];
    const float* mlp_b1     = (const float*)d_in[13];
    const float* mlp_W2     = (const float*)d_in[14];
    const float* mlp_b2     = (const float*)d_in[15];
    const float* mlp_W3     = (const float*)d_in[16];
    const float* mlp_b3     = (const float*)d_in[17];
    float* out = (float*)d_out;

    // Workspace layout (floats), ~62 MB total.
    float* ws = (float*)d_ws;
    const size_t NH = (size_t)N_NODES * HID;          // 5,120,000
    float* x     = ws;                                 // [N,256]
    float* h     = ws + NH;                            // [N,256] (also xpad home)
    float* msgw  = ws + 2 * NH;                        // [N,256]
    float* Wpad  = ws + 3 * NH;                        // [80,256]
    float* gmean = Wpad + (size_t)IN_PAD * HID;        // [G,256]
    float* h1    = gmean + (size_t)N_GRAPHS * HID;     // [G,1024]
    float* h2    = h1 + (size_t)N_GRAPHS * M1;         // [G,512]
    float* xpad  = h;  // reuse h region before it is first needed

    // 1) Pad inputs for the IN=74 -> 80 embedding GEMM.
    pad_x_kernel<<<ceil_div(N_NODES * IN_PAD, 256), 256, 0, stream>>>(node_h, xpad);
    pad_w_kernel<<<ceil_div(IN_PAD * HID, 256), 256, 0, stream>>>(node_emb_W, Wpad);

    // 2) x = node_h @ node_emb_W + node_emb_b       [20000,80]@[80,256]
    {
        int tiles = ceil_div(N_NODES, 64) * (HID / 16);
        wmma_gemm_kernel<<<ceil_div(tiles, 8), 256, 0, stream>>>(
            xpad, Wpad, node_emb_b, x, N_NODES, HID, IN_PAD, 0);
    }

    // 3) GCN layers
    for (int i = 0; i < NLAYERS; ++i) {
        int tiles = ceil_div(N_NODES, 64) * (HID / 16);
        // h = x @ emb_W[i] + emb_b[i]
        wmma_gemm_kernel<<<ceil_div(tiles, 8), 256, 0, stream>>>(
            x, emb_W + (size_t)i * HID * HID, emb_b + (size_t)i * HID,
            h, N_NODES, HID, HID, 0);
        // msgw = sw0 * relu(segsum(h[src])/deg + lb) + sw1 * x
        aggregate_kernel<<<N_NODES, 256, 0, stream>>>(
            h, x, src, dst, layer_bias + (size_t)i * HID, sum_w + 2 * i, msgw);
        // x = msgw @ trans_W[i] + trans_b[i]
        wmma_gemm_kernel<<<ceil_div(tiles, 8), 256, 0, stream>>>(
            msgw, trans_W + (size_t)i * HID * HID, trans_b + (size_t)i * HID,
            x, N_NODES, HID, HID, 0);
    }

    // 4) Per-graph mean readout
    graph_mean_kernel<<<N_GRAPHS, 256, 0, stream>>>(x, graph_id, gmean);

    // 5) MLP head
    {
        int tiles = ceil_div(N_GRAPHS, 64) * (M1 / 16);
        wmma_gemm_kernel<<<ceil_div(tiles, 8), 256, 0, stream>>>(
            gmean, mlp_W1, mlp_b1, h1, N_GRAPHS, M1, HID, 1);
    }
    {
        int tiles = ceil_div(N_GRAPHS, 64) * (M2 / 16);
        wmma_gemm_kernel<<<ceil_div(tiles, 8), 256, 0, stream>>>(
            h1, mlp_W2, mlp_b2, h2, N_GRAPHS, M2, M1, 1);
    }
    final_head_kernel<<<N_GRAPHS, 256, 0, stream>>>(h2, mlp_W3, mlp_b3, out);
}